// CustomMultiresLayer_81578608820280
// MI455X (gfx1250) — compile-verified
//
#include <hip/hip_runtime.h>

typedef __attribute__((ext_vector_type(2))) float v2f;
typedef __attribute__((ext_vector_type(8))) float v8f;

#define B_      2
#define D_      1024
#define L_      4096
#define N_      (B_ * L_)      /* 8192 columns (b,l) */
#define FS_     4
#define DEPTH_  11
#define LN_EPS  1e-5f
#define KB_     32             /* GEMM K block */
#define BPS_    (2 * 128 + 16) /* B LDS k-pair row stride (floats): 16-float pad
                                  -> pair-row +1 lands on banks 16..47 (conflict-free b64) */

// ---------------------------------------------------------------------------
// Kernel 1: dilated causal depthwise conv chain + sigmoid gating -> y[c][n]
// One workgroup per (b,d) row. Row kept in LDS; gating is elementwise so
// b_prev / y accumulate in registers (16 elements per thread).
// ---------------------------------------------------------------------------
__global__ __launch_bounds__(256) void mr_conv_gate(const float* __restrict__ x,
                                                    const float* __restrict__ h0,
                                                    const float* __restrict__ h1,
                                                    float* __restrict__ y_ws) {
    __shared__ float a_sh[L_];               // 16 KB
    const int row = blockIdx.x;              // row = b*D + d
    const int b   = row >> 10;
    const int d   = row & (D_ - 1);
    const int t   = threadIdx.x;

    const float* __restrict__ xrow = x + (size_t)row * L_;

    float f0[FS_], f1[FS_];
#pragma unroll
    for (int k = 0; k < FS_; ++k) { f0[k] = h0[d * FS_ + k]; f1[k] = h1[d * FS_ + k]; }

    float yacc[16], bprev[16];
#pragma unroll
    for (int j = 0; j < 16; ++j) {
        a_sh[j * 256 + t] = xrow[j * 256 + t];
        yacc[j] = 0.0f;
        bprev[j] = 0.0f;
    }
    __syncthreads();

    int dil = 1;
    for (int m = 0; m < DEPTH_; ++m) {
        float anew[16], bcur[16];
#pragma unroll
        for (int j = 0; j < 16; ++j) {
            const int l = j * 256 + t;
            float s0 = 0.0f, s1 = 0.0f;
#pragma unroll
            for (int k = 0; k < FS_; ++k) {
                const int idx = l - dil * (FS_ - 1 - k);
                const float av = (idx >= 0) ? a_sh[idx] : 0.0f;
                s0 += f0[k] * av;
                s1 += f1[k] * av;
            }
            anew[j] = s0;
            bcur[j] = s1;
        }
        if (m >= 1) {
            const float w = (m == 1) ? 2.0f : 1.0f;  // gated-carry quirk doubles m==1
#pragma unroll
            for (int j = 0; j < 16; ++j) {
                const float sg = 1.0f / (1.0f + __expf(-anew[j]));
                yacc[j] += w * sg * bprev[j];
            }
        }
#pragma unroll
        for (int j = 0; j < 16; ++j) bprev[j] = bcur[j];
        __syncthreads();
#pragma unroll
        for (int j = 0; j < 16; ++j) a_sh[j * 256 + t] = anew[j];
        __syncthreads();
        dil <<= 1;
    }

    float* __restrict__ yrow = y_ws + (size_t)d * N_ + (size_t)b * L_;
#pragma unroll
    for (int j = 0; j < 16; ++j) yrow[j * 256 + t] = yacc[j];
}

// ---------------------------------------------------------------------------
// Kernel 2: z[o][n] = sum_c W[o][c] * y[c][n] + b_mix[o] + x[b,o,l]
// fp32 WMMA (v_wmma_f32_16x16x4_f32), 64(M) x 128(N) tile per workgroup,
// 8 waves in a 4x2 grid, 4 accumulator tiles (16x64 strip) per wave.
// B tile stored k-pair interleaved so each B fragment is ONE aligned
// ds_load_b64 (no v_mov repacking around the WMMAs).
// ---------------------------------------------------------------------------
__global__ __launch_bounds__(256) void mix_gemm_wmma(const float* __restrict__ w_mix,
                                                     const float* __restrict__ y_ws,
                                                     const float* __restrict__ x,
                                                     const float* __restrict__ b_mix,
                                                     float* __restrict__ z_ws) {
    __shared__ float Asl[64][KB_ + 2];          // W tile, stride 34 (even -> v2f aligned)
    __shared__ float Bsl[(KB_ / 2) * BPS_];     // y tile, (k,n) -> [(k>>1)*BPS_ + 2n + (k&1)]

    const int t    = threadIdx.x;
    const int wv   = t >> 5;              // wave id 0..7 (wave32)
    const int lane = t & 31;
    const int half = lane >> 4;           // 0/1: which 16-lane half
    const int ln16 = lane & 15;
    const int wm   = wv & 3;              // M sub-tile (16 rows)
    const int wn   = wv >> 2;             // N sub-strip (64 cols)

    const int m0 = blockIdx.y * 64;
    const int n0 = blockIdx.x * 128;

    v8f acc[4] = {};

    for (int k0 = 0; k0 < D_; k0 += KB_) {
        // stage A (64x32), coalesced, conflict-free stores
#pragma unroll
        for (int i = 0; i < 8; ++i) {
            const int e = i * 256 + t;
            Asl[e >> 5][e & 31] = w_mix[(size_t)(m0 + (e >> 5)) * D_ + k0 + (e & 31)];
        }
        // stage B (32x128) into k-pair interleaved layout, coalesced global reads
#pragma unroll
        for (int i = 0; i < 16; ++i) {
            const int e = i * 256 + t;
            const int r = e >> 7;         // k within block (0..31)
            const int c = e & 127;        // n within tile
            Bsl[(r >> 1) * BPS_ + 2 * c + (r & 1)] =
                y_ws[(size_t)(k0 + r) * N_ + n0 + c];
        }
        __syncthreads();

        if (k0 + KB_ < D_) {  // warm next K block (global_prefetch_b8)
            __builtin_prefetch(&w_mix[(size_t)(m0 + (t >> 2)) * D_ + k0 + KB_ + 8 * (t & 3)], 0, 1);
            __builtin_prefetch(&y_ws[(size_t)(k0 + KB_ + (t >> 5)) * N_ + n0 + 16 * (t & 7)], 0, 1);
        }

#pragma unroll
        for (int kk = 0; kk < KB_; kk += 4) {
            // A frag (16x4 f32): lane m=ln16; VGPR0 = K=2h, VGPR1 = K=2h+1
            const v2f afrag = *(const v2f*)&Asl[wm * 16 + ln16][kk + 2 * half];
#pragma unroll
            for (int s = 0; s < 4; ++s) {
                const int nn = wn * 64 + s * 16 + ln16;
                // B frag rows kk+2h, kk+2h+1 at col nn: one contiguous 8B LDS load
                const v2f bfrag = *(const v2f*)&Bsl[(kk / 2 + half) * BPS_ + 2 * nn];
                acc[s] = __builtin_amdgcn_wmma_f32_16x16x4_f32(
                    false, afrag, false, bfrag, (short)0, acc[s], false, false);
            }
        }
        __syncthreads();
    }

    // epilogue: + b_mix[o] + x[b,o,l]; store z as [o][n]
#pragma unroll
    for (int s = 0; s < 4; ++s) {
        const int n  = n0 + wn * 64 + s * 16 + ln16;
        const int bb = n >> 12;          // n / L_
        const int l  = n & (L_ - 1);
#pragma unroll
        for (int i = 0; i < 8; ++i) {
            const int o = m0 + wm * 16 + i + 8 * half;   // C/D layout: VGPR i -> M=i / i+8
            const float v = acc[s][i] + b_mix[o] + x[(size_t)(bb * D_ + o) * L_ + l];
            z_ws[(size_t)o * N_ + n] = v;
        }
    }
}

// ---------------------------------------------------------------------------
// Kernel 3: LayerNorm over channels per column n, write out (B,D,L)
// 64 columns per workgroup; 4 threads/column reduce 256 channels each.
// ---------------------------------------------------------------------------
__global__ __launch_bounds__(256) void ln_kernel(const float* __restrict__ z_ws,
                                                 const float* __restrict__ gamma,
                                                 const float* __restrict__ beta,
                                                 float* __restrict__ out) {
    __shared__ float ssum[4][64];
    __shared__ float ssq[4][64];
    __shared__ float smu[64];
    __shared__ float srs[64];

    const int t   = threadIdx.x;
    const int j   = t & 63;
    const int seg = t >> 6;
    const int n   = blockIdx.x * 64 + j;

    float s = 0.0f, q = 0.0f;
    for (int oi = 0; oi < 256; ++oi) {
        const int o = seg * 256 + oi;
        const float v = z_ws[(size_t)o * N_ + n];
        s += v;
        q += v * v;
    }
    ssum[seg][j] = s;
    ssq[seg][j]  = q;
    __syncthreads();

    if (seg == 0) {
        const float S  = ssum[0][j] + ssum[1][j] + ssum[2][j] + ssum[3][j];
        const float Q  = ssq[0][j] + ssq[1][j] + ssq[2][j] + ssq[3][j];
        const float mu = S * (1.0f / (float)D_);
        const float va = Q * (1.0f / (float)D_) - mu * mu;
        smu[j] = mu;
        srs[j] = rsqrtf(va + LN_EPS);
    }
    __syncthreads();

    const float mu = smu[j];
    const float rs = srs[j];
    const int bb = n >> 12;
    const int l  = n & (L_ - 1);
    for (int oi = 0; oi < 256; ++oi) {
        const int o = seg * 256 + oi;
        const float v = z_ws[(size_t)o * N_ + n];
        out[(size_t)(bb * D_ + o) * L_ + l] = (v - mu) * rs * gamma[o] + beta[o];
    }
}

// ---------------------------------------------------------------------------
extern "C" void kernel_launch(void* const* d_in, const int* in_sizes, int n_in,
                              void* d_out, int out_size, void* d_ws, size_t ws_size,
                              hipStream_t stream) {
    (void)in_sizes; (void)n_in; (void)out_size; (void)ws_size;

    const float* x    = (const float*)d_in[0];
    const float* h0   = (const float*)d_in[1];
    const float* h1   = (const float*)d_in[2];
    const float* wmix = (const float*)d_in[3];
    const float* bmix = (const float*)d_in[4];
    const float* g    = (const float*)d_in[5];
    const float* bt   = (const float*)d_in[6];

    float* y_ws = (float*)d_ws;                       // [D_][N_]  32 MB
    float* z_ws = y_ws + (size_t)D_ * N_;             // [D_][N_]  32 MB

    mr_conv_gate<<<B_ * D_, 256, 0, stream>>>(x, h0, h1, y_ws);

    dim3 g2(N_ / 128, D_ / 64);                       // 64 x 16 workgroups
    mix_gemm_wmma<<<g2, 256, 0, stream>>>(wmix, y_ws, x, bmix, z_ws);

    ln_kernel<<<N_ / 64, 256, 0, stream>>>(z_ws, g, bt, (float*)d_out);
}